// TensorSerialDictatorship_78623671321181
// MI455X (gfx1250) — compile-verified
//
#include <hip/hip_runtime.h>

typedef __attribute__((ext_vector_type(2))) float v2f;
typedef __attribute__((ext_vector_type(8))) float v8f;

#define NW   96
#define NF   96
#define NP1  97               // options per agent (96 counterparts + unmatch)
#define ROWS (NW * NP1)       // 9312 flattened (agent, option) rows per tensor
#define TILES (ROWS / 16)     // 582 exactly
#define NAG  192

// ---------------------------------------------------------------------------
// Kernel 1: decode one-hot rankings to rank integers via WMMA.
//   rank[i][j] = sum_k k * X[i][j][k]   ==  (9312 x 97) matrix  times  kvec
// Done as D = A(16x4 f32) x B(4x16 f32) accumulated over 25 K-chunks, where
// B's 16 columns are all kvec (so D's columns are identical and the B lane
// layout is unambiguous). Also zeroes d_out.
// ---------------------------------------------------------------------------
__global__ __launch_bounds__(128)
void rank_extract_kernel(const float* __restrict__ W,
                         const float* __restrict__ F,
                         float* __restrict__ rankW,
                         float* __restrict__ rankF,
                         float* __restrict__ out)
{
    int gtid = blockIdx.x * 128 + threadIdx.x;
    if (gtid < NP1 * NP1) out[gtid] = 0.0f;   // zero M (EXEC restored before WMMA)

    int wg   = blockIdx.x * 4 + (threadIdx.x >> 5);   // global wave 0..1163
    int lane = threadIdx.x & 31;
    const float* src = (wg < TILES) ? W : F;
    float*       dst = (wg < TILES) ? rankW : rankF;
    int tile = (wg < TILES) ? wg : wg - TILES;
    int m0   = tile * 16;
    int half = lane >> 4;                 // 0: lanes 0-15 (K+0,K+1), 1: lanes 16-31 (K+2,K+3)
    int row  = m0 + (lane & 15);          // A: lanes 0-15 and 16-31 both hold M=0..15
    const float* rp = src + row * NP1;

    v8f c = {};
    for (int chunk = 0; chunk < 25; ++chunk) {
        int k0 = chunk * 4 + half * 2;
        int k1 = k0 + 1;
        // Branchless masking so EXEC stays all-1s for WMMA (K padded 97 -> 100).
        float w0 = (k0 < NP1) ? 1.0f : 0.0f;
        float w1 = (k1 < NP1) ? 1.0f : 0.0f;
        int ck0 = (k0 < NP1) ? k0 : (NP1 - 1);
        int ck1 = (k1 < NP1) ? k1 : (NP1 - 1);
        v2f a, b;
        a.x = rp[ck0] * w0;               // A[m, k0]
        a.y = rp[ck1] * w1;               // A[m, k1]
        b.x = (float)k0 * w0;             // B row k0 (constant across its 16 columns)
        b.y = (float)k1 * w1;             // B row k1
        c = __builtin_amdgcn_wmma_f32_16x16x4_f32(false, a, false, b,
                                                  (short)0, c, false, false);
    }
    // D columns are identical: lane 0 holds rows m0..m0+7, lane 16 rows m0+8..m0+15.
    if ((lane & 15) == 0) {
        int rbase = m0 + half * 8;
        for (int v = 0; v < 8; ++v) dst[rbase + v] = c[v];
    }
}

// ---------------------------------------------------------------------------
// Kernel 2: the inherently sequential serial-dictatorship loop (192 rounds).
// Single block; per round a 97-wide masked argmin via LDS tree reduction.
// ---------------------------------------------------------------------------
__global__ __launch_bounds__(128)
void dictatorship_kernel(const float* __restrict__ rankW,
                         const float* __restrict__ rankF,
                         const float* __restrict__ R,
                         float* __restrict__ out)
{
    __shared__ int sOrder[NAG];
    __shared__ int sAvailW[NP1];   // worker/firm still selectable (slot 96 = unmatch, always 1)
    __shared__ int sAvailF[NP1];
    __shared__ int sActW[NW];      // agent's ranking matrix not yet zeroed
    __shared__ int sActF[NF];
    __shared__ int sKey[128];

    int tid = threadIdx.x;
    // order[r] = agent a with R[a, r] == 1  (R = eye[perm], agent a dictates at perm[a])
    for (int r = tid; r < NAG; r += 128) {
        int acc = 0;
        for (int a = 0; a < NAG; ++a)
            if (R[a * NAG + r] > 0.5f) acc = a;
        sOrder[r] = acc;
    }
    if (tid < NP1) { sAvailW[tid] = 1; sAvailF[tid] = 1; }
    if (tid < NW)  { sActW[tid] = 1;  sActF[tid] = 1; }
    __syncthreads();

    for (int r = 0; r < NAG; ++r) {
        int ag   = sOrder[r];
        int isW  = (ag < NW) ? 1 : 0;
        int dict = isW ? ag : (ag - NW);
        const float* rrow = isW ? (rankW + dict * NP1) : (rankF + dict * NP1);
        const int*  avail = isW ? sAvailF : sAvailW;

        int key = 0x7FFFFFFF;
        if (tid < NP1 && avail[tid]) {
            int rk = (int)(rrow[tid] + 0.5f);          // exact small int from WMMA
            key = rk * 256 + tid;                      // rank-major key, idx in low bits
        }
        sKey[tid] = key;
        __syncthreads();
        for (int s = 64; s > 0; s >>= 1) {
            if (tid < s) {
                int o = sKey[tid + s];
                if (o < sKey[tid]) sKey[tid] = o;
            }
            __syncthreads();
        }
        if (tid == 0) {
            int act = isW ? sActW[dict] : sActF[dict];
            if (act) {
                int sel = sKey[0] & 255;               // first available choice
                if (isW) {
                    out[dict * NP1 + sel] = 1.0f;      // M[worker, firm/unmatch]
                    if (sel != NF) { sAvailF[sel] = 0; sActF[sel] = 0; }
                } else {
                    out[sel * NP1 + dict] = 1.0f;      // M[worker/unmatch, firm]
                    if (sel != NW) { sAvailW[sel] = 0; sActW[sel] = 0; }
                }
            }
            // dictator removed from the other side's menus unconditionally (dw1/df1 terms)
            if (isW) sAvailW[dict] = 0; else sAvailF[dict] = 0;
        }
        __syncthreads();
    }
}

// ---------------------------------------------------------------------------
extern "C" void kernel_launch(void* const* d_in, const int* in_sizes, int n_in,
                              void* d_out, int out_size, void* d_ws, size_t ws_size,
                              hipStream_t stream) {
    const float* W = (const float*)d_in[0];   // (96, 97, 97)
    const float* F = (const float*)d_in[1];   // (96, 97, 97)
    const float* R = (const float*)d_in[2];   // (192, 192)
    float* out   = (float*)d_out;             // (97, 97)
    float* rankW = (float*)d_ws;              // 9312 floats
    float* rankF = rankW + ROWS;              // 9312 floats

    rank_extract_kernel<<<dim3((TILES * 2) / 4), dim3(128), 0, stream>>>(
        W, F, rankW, rankF, out);
    dictatorship_kernel<<<dim3(1), dim3(128), 0, stream>>>(
        rankW, rankF, R, out);
}